// QuantGraphPoolOut_6906307412351
// MI455X (gfx1250) — compile-verified
//
#include <hip/hip_runtime.h>
#include <cstdint>
#include <cstddef>

// ---- problem constants (from reference) ----
#define POOL_SHIFT 2              // POOL_SIZE = 4
#define VGRID      64             // MAX_DIM / POOL_SIZE
#define NCHAN      64             // C
#define NUM_CELLS  (VGRID * VGRID * VGRID)   // 262144
// out_size = NUM_CELLS * NCHAN = 16,777,216 floats (64 MB -> L2 resident)

typedef float v4f __attribute__((ext_vector_type(4)));

// -------- kernel 1: init output to -inf sentinel (128-bit stores) --------
__global__ void qgp_init_kernel(float* __restrict__ out, int total4) {
    int t = blockIdx.x * blockDim.x + threadIdx.x;
    if (t >= total4) return;
    const float ninf = __builtin_inff() * -1.0f;
    v4f v = { ninf, ninf, ninf, ninf };
    ((v4f*)out)[t] = v;
}

// -------- kernel 2: scatter-max via CDNA5 FP32 max_num global atomics --------
// one thread per (point, 4-channel group): 16 threads per point
__global__ void qgp_scatter_kernel(const int*   __restrict__ verts,
                                   const float* __restrict__ feats,
                                   float*       __restrict__ out,
                                   int npoints) {
    int t = blockIdx.x * blockDim.x + threadIdx.x;
    int i = t >> 4;                 // point index
    int g = (t & 15) << 2;          // channel base (0,4,...,60)
    if (i >= npoints) return;

    // voxel cell index: (x/4 * G + y/4) * G + z/4
    int vx = verts[3 * i + 0] >> POOL_SHIFT;
    int vy = verts[3 * i + 1] >> POOL_SHIFT;
    int vz = verts[3 * i + 2] >> POOL_SHIFT;
    int cell = (vx * VGRID + vy) * VGRID + vz;

    // coalesced 128-bit feature load; non-temporal so the 256MB feature
    // stream does not evict the L2-resident 64MB output
    v4f f = __builtin_nontemporal_load((const v4f*)(feats + (size_t)i * NCHAN + g));

    float* base = out + (size_t)cell * NCHAN + g;

    // GLOBAL_ATOMIC_MAX_NUM_F32: IEEE maximumNumber RMW performed in L2
    // atomic units (no return -> STOREcnt, drained by s_endpgm implicit wait).
    asm volatile("global_atomic_max_num_f32 %0, %1, off scope:SCOPE_DEV"
                 :: "v"(base), "v"(f.x) : "memory");
    asm volatile("global_atomic_max_num_f32 %0, %1, off offset:4 scope:SCOPE_DEV"
                 :: "v"(base), "v"(f.y) : "memory");
    asm volatile("global_atomic_max_num_f32 %0, %1, off offset:8 scope:SCOPE_DEV"
                 :: "v"(base), "v"(f.z) : "memory");
    asm volatile("global_atomic_max_num_f32 %0, %1, off offset:12 scope:SCOPE_DEV"
                 :: "v"(base), "v"(f.w) : "memory");
}

// -------- kernel 3: untouched cells (-inf sentinel) -> 0 --------
__global__ void qgp_finalize_kernel(float* __restrict__ out, int total4) {
    int t = blockIdx.x * blockDim.x + threadIdx.x;
    if (t >= total4) return;
    const float ninf = __builtin_inff() * -1.0f;
    v4f v = ((v4f*)out)[t];
    v.x = (v.x == ninf) ? 0.0f : v.x;
    v.y = (v.y == ninf) ? 0.0f : v.y;
    v.z = (v.z == ninf) ? 0.0f : v.z;
    v.w = (v.w == ninf) ? 0.0f : v.w;
    ((v4f*)out)[t] = v;
}

extern "C" void kernel_launch(void* const* d_in, const int* in_sizes, int n_in,
                              void* d_out, int out_size, void* d_ws, size_t ws_size,
                              hipStream_t stream) {
    const int*   verts = (const int*)  d_in[0];   // [N,3] int32
    const float* feats = (const float*)d_in[1];   // [N,C] float32
    float*       out   = (float*)d_out;           // [NUM_CELLS*C] float32

    const int npoints = in_sizes[0] / 3;          // 1,000,000
    const int total4  = out_size / 4;             // 4,194,304 float4 groups

    const int BLK = 256;                          // 8 waves (wave32)

    // 1) init to -inf sentinel
    qgp_init_kernel<<<(total4 + BLK - 1) / BLK, BLK, 0, stream>>>(out, total4);

    // 2) scatter-max: N*16 threads (one per point x 4-channel group)
    long long nthreads = (long long)npoints * 16;
    int nblocks = (int)((nthreads + BLK - 1) / BLK);
    qgp_scatter_kernel<<<nblocks, BLK, 0, stream>>>(verts, feats, out, npoints);

    // 3) sentinel -> 0 for empty cells
    qgp_finalize_kernel<<<(total4 + BLK - 1) / BLK, BLK, 0, stream>>>(out, total4);
}